// IntraContrastiveLossDNS_14491219657441
// MI455X (gfx1250) — compile-verified
//
#include <hip/hip_runtime.h>
#include <math.h>

typedef __attribute__((ext_vector_type(2))) float v2f;
typedef __attribute__((ext_vector_type(8))) float v8f;

#define NN   64
#define NN2  4096           // N*N
#define PP   2080           // N*(N+1)/2
#define SS   32
#define MM   128
#define BB   16
#define CC   256
#define BP   33280          // BB*PP
#define KK   2
#define EE   2048
#define NEGS 512

// ---------------------------------------------------------------- threefry2x32
__device__ __forceinline__ unsigned rotl32(unsigned x, int r) {
    return (x << r) | (x >> (32 - r));
}

__device__ __forceinline__ void threefry2x32(unsigned k0, unsigned k1,
                                             unsigned c0, unsigned c1,
                                             unsigned& o0, unsigned& o1) {
    unsigned ks2 = 0x1BD11BDAu ^ k0 ^ k1;
    unsigned x0 = c0 + k0, x1 = c1 + k1;
    const int r1[4] = {13, 15, 26, 6};
    const int r2[4] = {17, 29, 16, 24};
    #pragma unroll
    for (int i = 0; i < 4; ++i) { x0 += x1; x1 = rotl32(x1, r1[i]); x1 ^= x0; }
    x0 += k1;  x1 += ks2 + 1u;
    #pragma unroll
    for (int i = 0; i < 4; ++i) { x0 += x1; x1 = rotl32(x1, r2[i]); x1 ^= x0; }
    x0 += ks2; x1 += k0 + 2u;
    #pragma unroll
    for (int i = 0; i < 4; ++i) { x0 += x1; x1 = rotl32(x1, r1[i]); x1 ^= x0; }
    x0 += k0;  x1 += k1 + 3u;
    #pragma unroll
    for (int i = 0; i < 4; ++i) { x0 += x1; x1 = rotl32(x1, r2[i]); x1 ^= x0; }
    x0 += k1;  x1 += ks2 + 4u;
    #pragma unroll
    for (int i = 0; i < 4; ++i) { x0 += x1; x1 = rotl32(x1, r1[i]); x1 ^= x0; }
    x0 += ks2; x1 += k0 + 5u;
    o0 = x0; o1 = x1;
}

// ---------------------------------------------------------------- tiny setup
__global__ void flatpos_kernel(int* __restrict__ flat_pos) {
    int i = threadIdx.x;                    // row of triu, 64 threads
    if (i >= NN) return;
    int start = i * NN - (i * (i - 1)) / 2; // entries before row i
    for (int t = 0; t < NN - i; ++t) flat_pos[start + t] = i * NN + i + t;
}

__global__ void scatter_kernel(const int* __restrict__ num_sentences,
                               const int* __restrict__ num_targets,
                               int* __restrict__ m2s, int* __restrict__ s2b,
                               int* __restrict__ b2s, int* __restrict__ sstart) {
    if (threadIdx.x != 0 || blockIdx.x != 0) return;
    int idx = 0;
    for (int s = 0; s < SS && idx < MM; ++s)
        for (int t = 0; t < num_targets[s] && idx < MM; ++t) m2s[idx++] = s;
    while (idx < MM) m2s[idx++] = SS - 1;
    idx = 0;
    for (int b = 0; b < BB && idx < SS; ++b)
        for (int t = 0; t < num_sentences[b] && idx < SS; ++t) s2b[idx++] = b;
    while (idx < SS) s2b[idx++] = BB - 1;
    int c = 0;
    for (int b = 0; b < BB; ++b) { b2s[b] = c; c += num_sentences[b]; }
    c = 0;
    for (int s = 0; s < SS; ++s) { sstart[s] = c; c += num_targets[s]; }
}

// top-2 proposals per moment by iou2ds (ties -> lower index, scan ascending)
__global__ void topk2_kernel(const float* __restrict__ iou2ds,
                             const int* __restrict__ flat_pos,
                             int* __restrict__ topk) {
    int m = blockIdx.x * blockDim.x + threadIdx.x;
    if (m >= MM) return;
    const float* base = iou2ds + (size_t)m * NN2;
    float v1 = -INFINITY, v2 = -INFINITY;
    int i1 = 0, i2 = 0;
    for (int p = 0; p < PP; ++p) {
        float v = base[flat_pos[p]];
        if (v > v1)      { v2 = v1; i2 = i1; v1 = v; i1 = p; }
        else if (v > v2) { v2 = v;  i2 = p; }
    }
    topk[m * 2 + 0] = i1;
    topk[m * 2 + 1] = i2;
}

// ---------------------------------------------------------------- normalization
// One wave per output row: gather 256 channels (stride N*N) and L2-normalize.
__global__ void posvf_kernel(const float* __restrict__ vfeat,
                             const int* __restrict__ flat_pos,
                             const int* __restrict__ m2s,
                             const int* __restrict__ topk,
                             float* __restrict__ posvf) {
    const int wave = (blockIdx.x * blockDim.x + threadIdx.x) >> 5; // 0..255
    const int lane = threadIdx.x & 31;
    const int m = wave >> 1;
    const int s = m2s[m];
    const int p = topk[wave];
    const float* src = vfeat + (size_t)s * CC * NN2 + flat_pos[p];
    float x[8]; float ss = 0.f;
    #pragma unroll
    for (int i = 0; i < 8; ++i) {
        x[i] = src[(size_t)(lane + 32 * i) * NN2];
        ss += x[i] * x[i];
    }
    for (int off = 16; off > 0; off >>= 1) ss += __shfl_xor(ss, off, 32);
    float denom = fmaxf(sqrtf(ss), 1e-12f);
    float* dst = posvf + (size_t)wave * CC;
    #pragma unroll
    for (int i = 0; i < 8; ++i) dst[lane + 32 * i] = x[i] / denom;
}

__global__ void bank_kernel(const float* __restrict__ vfeat,
                            const int* __restrict__ flat_pos,
                            const int* __restrict__ b2s,
                            float* __restrict__ bank) {
    const int wave = (blockIdx.x * blockDim.x + threadIdx.x) >> 5; // 0..BP-1
    const int lane = threadIdx.x & 31;
    const int b = wave / PP;
    const int p = wave - b * PP;
    const int s = b2s[b];
    const float* src = vfeat + (size_t)s * CC * NN2 + flat_pos[p];
    float x[8]; float ss = 0.f;
    #pragma unroll
    for (int i = 0; i < 8; ++i) {
        x[i] = src[(size_t)(lane + 32 * i) * NN2];
        ss += x[i] * x[i];
    }
    for (int off = 16; off > 0; off >>= 1) ss += __shfl_xor(ss, off, 32);
    float denom = fmaxf(sqrtf(ss), 1e-12f);
    float* dst = bank + (size_t)wave * CC;
    #pragma unroll
    for (int i = 0; i < 8; ++i) dst[lane + 32 * i] = x[i] / denom;
}

// ---------------------------------------------------------------- WMMA GEMM
// D(256 x BP) = A(256 x 256) * bank^T  using V_WMMA_F32_16X16X4_F32.
// One wave -> 16-row x 64-col tile (4 accumulators), 64 K-steps of 4.
// A frag: a[v] = A[lane&15][k + (lane>>4)*2 + v]
// B frag: b[v] = bank[n + lane&15][k + (lane>>4)*2 + v]   (B = 4x16 K-major)
// D frag: c[r] = D[rowTile*16 + r + (lane>>4)*8][n + lane&15]
__global__ void wmma_gemm_kernel(const float* __restrict__ A,
                                 const float* __restrict__ Bm,
                                 float* __restrict__ D) {
    const int wave = (blockIdx.x * blockDim.x + threadIdx.x) >> 5;
    const int lane = threadIdx.x & 31;
    const int rowTile = wave / 520;             // 16 row tiles
    const int colGrp  = wave - rowTile * 520;   // 520 groups of 4 col tiles
    const int n0 = colGrp * 64;
    const int lr = lane & 15;
    const int kh = (lane >> 4) * 2;
    const float* arow = A  + (size_t)(rowTile * 16 + lr) * CC + kh;
    const float* b0   = Bm + (size_t)(n0 +  0 + lr) * CC + kh;
    const float* b1   = Bm + (size_t)(n0 + 16 + lr) * CC + kh;
    const float* b2   = Bm + (size_t)(n0 + 32 + lr) * CC + kh;
    const float* b3   = Bm + (size_t)(n0 + 48 + lr) * CC + kh;
    v8f c0 = {}, c1 = {}, c2 = {}, c3 = {};
    #pragma unroll 4
    for (int k = 0; k < CC; k += 4) {
        v2f a;  a.x  = arow[k]; a.y  = arow[k + 1];
        v2f x0; x0.x = b0[k];   x0.y = b0[k + 1];
        v2f x1; x1.x = b1[k];   x1.y = b1[k + 1];
        v2f x2; x2.x = b2[k];   x2.y = b2[k + 1];
        v2f x3; x3.x = b3[k];   x3.y = b3[k + 1];
        c0 = __builtin_amdgcn_wmma_f32_16x16x4_f32(false, a, false, x0, (short)0, c0, false, false);
        c1 = __builtin_amdgcn_wmma_f32_16x16x4_f32(false, a, false, x1, (short)0, c1, false, false);
        c2 = __builtin_amdgcn_wmma_f32_16x16x4_f32(false, a, false, x2, (short)0, c2, false, false);
        c3 = __builtin_amdgcn_wmma_f32_16x16x4_f32(false, a, false, x3, (short)0, c3, false, false);
    }
    const int rbase = rowTile * 16 + (lane >> 4) * 8;
    #pragma unroll
    for (int r = 0; r < 8; ++r) {
        D[(size_t)(rbase + r) * BP + n0 +  0 + lr] = c0[r];
        D[(size_t)(rbase + r) * BP + n0 + 16 + lr] = c1[r];
        D[(size_t)(rbase + r) * BP + n0 + 32 + lr] = c2[r];
        D[(size_t)(rbase + r) * BP + n0 + 48 + lr] = c3[r];
    }
}

// ---------------------------------------------------------------- DNS keys
// fused similarity, pos-mask, gumbel-perturbed log-weight -> ordered u32 key.
__global__ void fused_keys_kernel(const float* __restrict__ intra,
                                  const float* __restrict__ iou2d,
                                  const int* __restrict__ flat_pos,
                                  const int* __restrict__ s2b,
                                  const int* __restrict__ sstart,
                                  const int* __restrict__ num_targets,
                                  unsigned* __restrict__ keys,
                                  float* __restrict__ negmask) {
    const int idx = blockIdx.x * blockDim.x + threadIdx.x; // 0..SS*BP-1
    const int s = idx / BP;
    const int j = idx - s * BP;
    const int nt = num_targets[s];
    const int m0 = sstart[s];
    float acc = 0.f;
    for (int t = 0; t < nt; ++t) {
        int m = m0 + t;
        float r0 = intra[(size_t)(m * 2 + 0) * BP + j];
        float r1 = intra[(size_t)(m * 2 + 1) * BP + j];
        acc += ((r0 + r1) * 0.5f + 1.0f) * 0.5f;   // sim per moment
    }
    float fused = acc / (float)nt;
    const int b = j / PP;
    const int p = j - b * PP;
    bool pos = (s2b[s] == b) && (iou2d[(size_t)s * NN2 + flat_pos[p]] > 0.5f);
    negmask[idx] = pos ? 0.0f : 1.0f;              // fallback when epoch < start
    float w = pos ? 0.0f : fused * fused;
    float logw = (w > 0.0f) ? logf(fmaxf(w, 1e-30f)) : -INFINITY;
    // jax.random.uniform(key(42), (SS,BP), 1e-7, 1-1e-7) via threefry2x32
    const unsigned h = (SS * BP) / 2u;
    unsigned o0, o1, r;
    if ((unsigned)idx < h) { threefry2x32(0u, 42u, (unsigned)idx, (unsigned)idx + h, o0, o1); r = o0; }
    else                   { threefry2x32(0u, 42u, (unsigned)idx - h, (unsigned)idx, o0, o1); r = o1; }
    float u01 = __uint_as_float(0x3F800000u | (r >> 9)) - 1.0f;
    const float mn = 1e-7f;
    const float mx = (float)(1.0 - 1e-7);
    float u = fmaxf(mn, u01 * (mx - mn) + mn);
    float g = -logf(-logf(u));
    float score = logw + g;
    unsigned bits = __float_as_uint(score);
    keys[idx] = (bits & 0x80000000u) ? ~bits : (bits | 0x80000000u);
}

// exact top-512 per sentence: binary search on u32 keys + index-ordered ties
__global__ void select_neg_kernel(const unsigned* __restrict__ keys,
                                  const int* __restrict__ epoch,
                                  float* __restrict__ negmask) {
    if (*epoch < 1) return;                        // START_DNS_EPOCH
    const int s = blockIdx.x;
    const int t = threadIdx.x;
    const unsigned* row = keys + (size_t)s * BP;
    float* mrow = negmask + (size_t)s * BP;
    const int CH = (BP + 255) / 256;               // 130 contiguous per thread
    const int j0 = t * CH;
    const int j1 = (j0 + CH < BP) ? (j0 + CH) : BP;
    __shared__ int sh[256];
    __shared__ int sh2[256];
    __shared__ int sh_totgt;
    unsigned lo = 0u, hi = 0xFFFFFFFFu;
    while (lo < hi) {
        unsigned mid = (unsigned)((((unsigned long long)lo + (unsigned long long)hi) + 1ull) >> 1);
        int c = 0;
        for (int j = j0; j < j1; ++j) c += (row[j] >= mid) ? 1 : 0;
        sh[t] = c;
        __syncthreads();
        for (int off = 128; off > 0; off >>= 1) {
            if (t < off) sh[t] += sh[t + off];
            __syncthreads();
        }
        int total = sh[0];
        __syncthreads();
        if (total >= NEGS) lo = mid; else hi = mid - 1;
    }
    const unsigned kth = lo;                       // value of 512th largest
    int cg = 0, ce = 0;
    for (int j = j0; j < j1; ++j) {
        unsigned v = row[j];
        cg += (v > kth) ? 1 : 0;
        ce += (v == kth) ? 1 : 0;
    }
    sh[t] = cg; sh2[t] = ce;
    __syncthreads();
    if (t == 0) {
        int tot = 0;
        for (int i = 0; i < 256; ++i) tot += sh[i];
        int run = 0;
        for (int i = 0; i < 256; ++i) { int tmp = sh2[i]; sh2[i] = run; run += tmp; }
        sh_totgt = tot;
    }
    __syncthreads();
    const int need_eq = NEGS - sh_totgt;
    int rank = sh2[t];
    for (int j = j0; j < j1; ++j) {
        unsigned v = row[j];
        float m;
        if (v > kth)       m = 1.0f;
        else if (v == kth) { m = (rank < need_eq) ? 1.0f : 0.0f; ++rank; }
        else               m = 0.0f;
        mrow[j] = m;
    }
}

// ---------------------------------------------------------------- reductions
__global__ void ns_kernel(const float* __restrict__ intra,
                          const float* __restrict__ negmask,
                          const int* __restrict__ m2s,
                          float* __restrict__ NS) {
    const int r = blockIdx.x;                      // pos_vf row 0..255
    const int t = threadIdx.x;
    const int s = m2s[r >> 1];
    const float* irow = intra + (size_t)r * BP;
    const float* mrow = negmask + (size_t)s * BP;
    float acc = 0.f;
    for (int j = t; j < BP; j += 256) acc += expf(irow[j] / 0.1f) * mrow[j];
    __shared__ float red[256];
    red[t] = acc;
    __syncthreads();
    for (int off = 128; off > 0; off >>= 1) {
        if (t < off) red[t] += red[t + off];
        __syncthreads();
    }
    if (t == 0) NS[r] = red[0];
}

__global__ void loss_kernel(const float* __restrict__ posvf,
                            const float* __restrict__ NS,
                            float* __restrict__ out) {
    const int t = threadIdx.x;
    float acc = 0.f;
    for (int e = t; e < EE; e += 256) {
        int s = e >> 6;                 // sentence
        int q = e & 63;
        int rr = s * 8 + (q >> 3);      // ref row
        int pr = s * 8 + (q & 7);       // pos row
        const float* a = posvf + (size_t)rr * CC;
        const float* b = posvf + (size_t)pr * CC;
        float d = 0.f;
        for (int c = 0; c < CC; ++c) d += a[c] * b[c];
        float pl = (d - 0.0f) / 0.1f;   // (intra_pos - MARGIN) / T
        acc += -(pl - logf(expf(pl) + NS[rr]));
    }
    __shared__ float red[256];
    red[t] = acc;
    __syncthreads();
    for (int off = 128; off > 0; off >>= 1) {
        if (t < off) red[t] += red[t + off];
        __syncthreads();
    }
    if (t == 0) {
        float loss = red[0] / (float)EE;
        out[0] = loss * 1.0f;           // WEIGHT = 1.0
        out[1] = loss;
    }
}

// ---------------------------------------------------------------- launch
extern "C" void kernel_launch(void* const* d_in, const int* in_sizes, int n_in,
                              void* d_out, int out_size, void* d_ws, size_t ws_size,
                              hipStream_t stream) {
    const float* video_feats   = (const float*)d_in[0];
    const int*   num_sentences = (const int*)d_in[2];
    const int*   num_targets   = (const int*)d_in[3];
    const float* iou2d         = (const float*)d_in[4];
    const float* iou2ds        = (const float*)d_in[5];
    const int*   epoch         = (const int*)d_in[7];
    (void)in_sizes; (void)n_in; (void)out_size;

    char* w = (char*)d_ws;
    float*    bank     = (float*)w;    w += (size_t)BP * CC * 4;        // 34.1 MB
    float*    intra    = (float*)w;    w += (size_t)(MM * KK) * BP * 4; // 34.1 MB
    unsigned* keys     = (unsigned*)w; w += (size_t)SS * BP * 4;        // 4.26 MB
    float*    negmask  = (float*)w;    w += (size_t)SS * BP * 4;        // 4.26 MB
    float*    posvf    = (float*)w;    w += (size_t)(MM * KK) * CC * 4; // 256 KB
    int*      flat_pos = (int*)w;      w += (size_t)PP * 4;
    int*      topk     = (int*)w;      w += (size_t)MM * KK * 4;
    int*      m2s      = (int*)w;      w += (size_t)MM * 4;
    int*      s2b      = (int*)w;      w += (size_t)SS * 4;
    int*      b2s      = (int*)w;      w += (size_t)BB * 4;
    int*      sstart   = (int*)w;      w += (size_t)SS * 4;
    float*    NS       = (float*)w;    w += (size_t)(MM * KK) * 4;
    if ((size_t)(w - (char*)d_ws) > ws_size) return;   // workspace too small

    flatpos_kernel<<<1, 64, 0, stream>>>(flat_pos);
    scatter_kernel<<<1, 1, 0, stream>>>(num_sentences, num_targets, m2s, s2b, b2s, sstart);
    topk2_kernel<<<1, 128, 0, stream>>>(iou2ds, flat_pos, topk);
    posvf_kernel<<<32, 256, 0, stream>>>(video_feats, flat_pos, m2s, topk, posvf);
    bank_kernel<<<BP / 8, 256, 0, stream>>>(video_feats, flat_pos, b2s, bank);
    wmma_gemm_kernel<<<(16 * 520) / 8, 256, 0, stream>>>(posvf, bank, intra);
    fused_keys_kernel<<<(SS * BP) / 256, 256, 0, stream>>>(intra, iou2d, flat_pos,
                                                           s2b, sstart, num_targets,
                                                           keys, negmask);
    select_neg_kernel<<<SS, 256, 0, stream>>>(keys, epoch, negmask);
    ns_kernel<<<MM * KK, 256, 0, stream>>>(intra, negmask, m2s, NS);
    loss_kernel<<<1, 256, 0, stream>>>(posvf, NS, (float*)d_out);
}